// MultiOrdinalCutoffs_13511967113827
// MI455X (gfx1250) — compile-verified
//
#include <hip/hip_runtime.h>
#include <hip/hip_bf16.h>
#include <stdint.h>

// ext-vector types matching the probed amdgpu-toolchain (clang-23) signature of
// __builtin_amdgcn_tensor_load_to_lds:
//   (uint32x4 g0, int32x8 g1, int32x4 g2, int32x4 g3, int32x8 extra, i32 cpol)
typedef __attribute__((ext_vector_type(4))) unsigned int v4u;
typedef __attribute__((ext_vector_type(8))) int          v8i;
typedef __attribute__((ext_vector_type(4))) int          v4i;

#define ROWS_PER_BLOCK 128   // rows of the [B, Cmax] problem handled per block
#define BLOCK_THREADS  256   // 8 wave32 per block
#define CPAD           64    // virtual row width (Cmax <= 64); wave32-aligned

// diff[b,c] = (c < len[t]) ? input[b] - weights[t,c] : 0
// pos [b,c] = weights[t,c]        (padding already zeroed in the table)
__global__ __launch_bounds__(BLOCK_THREADS)
void mo_cutoffs_kernel(const float* __restrict__ input,    // [B]
                       const int*   __restrict__ ids,      // [B]
                       const float* __restrict__ weights,  // [T*Cmax]
                       const int*   __restrict__ lengths,  // [T]
                       float* __restrict__ out_diff,       // [B*Cmax]
                       float* __restrict__ out_pos,        // [B*Cmax]
                       int B, int T, int Cmax)
{
    __shared__ float s_w[16 * 63];            // cutoff table (<= 4032 B)
    __shared__ int   s_len[16];
    __shared__ float s_x[ROWS_PER_BLOCK];
    __shared__ int   s_id[ROWS_PER_BLOCK];

    const int tid = threadIdx.x;
    const int b0  = blockIdx.x * ROWS_PER_BLOCK;
    const int nW  = T * Cmax;                 // 1008 elements

    // ---- Stage the weights table into LDS with the Tensor Data Mover ----
#if __has_builtin(__builtin_amdgcn_tensor_load_to_lds) && __has_builtin(__builtin_amdgcn_s_wait_tensorcnt)
    if (tid == 0) {                           // wave 0 issues the DMA (EXEC ignored by TDM)
        const uint64_t ga  = (uint64_t)(uintptr_t)weights;
        // generic LDS pointer: low 32 bits are the LDS byte offset (aperture decode)
        const unsigned lds = (unsigned)(uintptr_t)(&s_w[0]);
        const unsigned n   = (unsigned)nW;    // elements, data_size = 4 B

        v4u g0;
        g0.x = 1u;                                      // count=1, user descriptor
        g0.y = lds;                                     // lds_addr
        g0.z = (unsigned)(ga & 0xffffffffu);            // global_addr[31:0]
        g0.w = (unsigned)((ga >> 32) & 0x01ffffffu)     // global_addr[56:32]
             | 0x80000000u;                             // type = 2 ("image")

        v8i g1;
        g1[0] = 0x00020000;                             // data_size = 2 -> 4 bytes
        g1[1] = (int)((n & 0xffffu) << 16);             // tensor_dim0[15:0]
        g1[2] = (int)((n >> 16) & 0xffffu)              // tensor_dim0[31:16]
              | (1 << 16);                              // tensor_dim1 = 1
        g1[3] = (int)((n & 0xffffu) << 16);             // tile_dim0 = n  (1-D tile)
        g1[4] = 1;                                      // tile_dim1 = 1
        g1[5] = (int)n;                                 // tensor_dim0_stride
        g1[6] = 0;
        g1[7] = 0;
        v4i g2 = {0, 0, 0, 0};                          // dims 3..5 unused
        v4i g3 = {0, 0, 0, 0};
        v8i gx = {0, 0, 0, 0, 0, 0, 0, 0};              // extra group (6-arg form)

        __builtin_amdgcn_tensor_load_to_lds(g0, g1, g2, g3, gx, /*cpol=*/0);
        __builtin_amdgcn_s_wait_tensorcnt(0);           // s_wait_tensorcnt 0
    }
#else
    for (int i = tid; i < nW; i += BLOCK_THREADS) s_w[i] = weights[i];
#endif

    // ---- Stage this block's scalars (coalesced) while the DMA runs ----
    if (tid < T) s_len[tid] = lengths[tid];
    for (int r = tid; r < ROWS_PER_BLOCK; r += BLOCK_THREADS) {
        const int b = b0 + r;
        if (b < B) { s_x[r] = input[b]; s_id[r] = ids[b]; }
    }
    __syncthreads();

    // ---- Main loop: lanes 0..63 cover one row; stores fully coalesced ----
    const int c  = tid & (CPAD - 1);          // column within row
    const int r0 = tid >> 6;                  // 0..3: row sub-group
    for (int r = r0; r < ROWS_PER_BLOCK; r += BLOCK_THREADS / CPAD) {
        const int b = b0 + r;
        if (b >= B) break;
        const int   t = s_id[r];              // LDS broadcast
        const float x = s_x[r];               // LDS broadcast
        if (c < Cmax) {
            const float w = s_w[t * Cmax + c];            // stride-1, conflict-free
            const float d = (c < s_len[t]) ? (x - w) : 0.0f;
            const size_t o = (size_t)b * (size_t)Cmax + (size_t)c;
            out_diff[o] = d;                  // coalesced b32 stores
            out_pos[o]  = w;
        }
    }
}

extern "C" void kernel_launch(void* const* d_in, const int* in_sizes, int n_in,
                              void* d_out, int out_size, void* d_ws, size_t ws_size,
                              hipStream_t stream) {
    const float* input   = (const float*)d_in[0];   // [B,1] float32
    const int*   ids     = (const int*)  d_in[1];   // [B]   int32
    const float* weights = (const float*)d_in[2];   // [T,Cmax] float32
    const int*   lengths = (const int*)  d_in[3];   // [T]   int32

    const int B    = in_sizes[1];
    const int T    = in_sizes[3];
    const int Cmax = in_sizes[2] / T;

    float* out      = (float*)d_out;
    float* out_diff = out;                               // [B*Cmax]
    float* out_pos  = out + (size_t)B * (size_t)Cmax;    // [B*Cmax]

    const int grid = (B + ROWS_PER_BLOCK - 1) / ROWS_PER_BLOCK;
    hipLaunchKernelGGL(mo_cutoffs_kernel, dim3(grid), dim3(BLOCK_THREADS), 0, stream,
                       input, ids, weights, lengths, out_diff, out_pos, B, T, Cmax);
}